// RelationalGraphConvolution_1297080124152
// MI455X (gfx1250) — compile-verified
//
#include <hip/hip_runtime.h>
#include <hip/hip_bf16.h>

#define DIM 256

typedef __attribute__((ext_vector_type(2))) float v2f;
typedef __attribute__((ext_vector_type(8))) float v8f;

// ---------------------------------------------------------------------------
// Zero-fill (float4 vectorized).
// ---------------------------------------------------------------------------
__global__ void zero_kernel(float4* __restrict__ p, long n4) {
    long i = (long)blockIdx.x * blockDim.x + threadIdx.x;
    if (i < n4) p[i] = make_float4(0.f, 0.f, 0.f, 0.f);
}

// ---------------------------------------------------------------------------
// H = A @ W  with fp32 WMMA (V_WMMA_F32_16X16X4_F32).
// One wave computes a 16x128 output stripe: 8 accumulators of 16x16
// (64 VGPRs of C/D -> no spilling). blockIdx.y in {0,1} picks column half.
// blockDim.x = 64 (2 waves); gridDim = (n_rows/32, 2).
//
// A-fragment (16x4 f32, 2 VGPRs/lane):
//   lanes 0-15 : M = lane,    v0 = K=k+0, v1 = K=k+1
//   lanes 16-31: M = lane-16, v0 = K=k+2, v1 = K=k+3
// B-fragment (4x16 f32, mirrored):
//   lanes 0-15 : N = lane,    v0 = row k+0, v1 = row k+1
//   lanes 16-31: N = lane-16, v0 = row k+2, v1 = row k+3
// C/D (16x16 f32, 8 VGPRs): vgpr i: lanes 0-15 -> M=i, lanes 16-31 -> M=8+i,
//   N = lane & 15.
// ---------------------------------------------------------------------------
__global__ __launch_bounds__(64)
void gemm_f32_wmma_kernel(const float* __restrict__ A,
                          const float* __restrict__ W,
                          float* __restrict__ H) {
    const int wave = threadIdx.x >> 5;
    const int lane = threadIdx.x & 31;
    const int tile = blockIdx.x * 2 + wave;      // 16-row tile index
    const int row0 = tile * 16;
    const int col0 = blockIdx.y * 128;           // column half
    const int m    = lane & 15;
    const int kh   = lane >> 4;                  // 0 or 1 -> K offset 2*kh

    v8f zero = {};
    v8f acc[8];
#pragma unroll
    for (int t = 0; t < 8; ++t) acc[t] = zero;

    const float* arow = A + (size_t)(row0 + m) * DIM;

    for (int k = 0; k < DIM; k += 4) {
        // A fragment: 2 consecutive floats per lane (8-byte aligned)
        v2f afrag = *(const v2f*)(arow + k + 2 * kh);
        const float* wrow = W + (size_t)(k + 2 * kh) * DIM + col0 + m;
#pragma unroll
        for (int nt = 0; nt < 8; ++nt) {
            v2f bfrag;
            bfrag.x = wrow[nt * 16];
            bfrag.y = wrow[nt * 16 + DIM];
            acc[nt] = __builtin_amdgcn_wmma_f32_16x16x4_f32(
                false, afrag, false, bfrag, (short)0, acc[nt], false, false);
        }
    }

#pragma unroll
    for (int nt = 0; nt < 8; ++nt) {
#pragma unroll
        for (int i = 0; i < 8; ++i) {
            int mm = i + 8 * kh;
            H[(size_t)(row0 + mm) * DIM + col0 + nt * 16 + m] = acc[nt][i];
        }
    }
}

// ---------------------------------------------------------------------------
// COO SpMM scatter: out[row[e]] += val[e] * h[col[e]]   (atomic f32)
// One wave per edge; each lane moves 2 float4 (= 8 floats) of the 256-wide row.
// ---------------------------------------------------------------------------
__global__ void spmm_atomic_kernel(const float* __restrict__ val,
                                   const int* __restrict__ row,
                                   const int* __restrict__ col,
                                   const float* __restrict__ h,
                                   float* __restrict__ out,
                                   int nedges) {
    const int lane = threadIdx.x & 31;
    const int e = blockIdx.x * (blockDim.x >> 5) + (threadIdx.x >> 5);
    if (e >= nedges) return;

    const int   r = row[e];
    const int   c = col[e];
    const float v = val[e];

    const float4* src = (const float4*)(h + (size_t)c * DIM);
    float*        dst = out + (size_t)r * DIM;

#pragma unroll
    for (int it = 0; it < 2; ++it) {
        const int q = lane + it * 32;            // float4 index 0..63
        float4 d = src[q];
        float* p = dst + q * 4;
#if defined(__HIP_DEVICE_COMPILE__)
        unsafeAtomicAdd(p + 0, v * d.x);         // -> global_atomic_add_f32
        unsafeAtomicAdd(p + 1, v * d.y);
        unsafeAtomicAdd(p + 2, v * d.z);
        unsafeAtomicAdd(p + 3, v * d.w);
#else
        atomicAdd(p + 0, v * d.x);
        atomicAdd(p + 1, v * d.y);
        atomicAdd(p + 2, v * d.z);
        atomicAdd(p + 3, v * d.w);
#endif
    }
}

// ---------------------------------------------------------------------------
// In-place ReLU (float4 vectorized)
// ---------------------------------------------------------------------------
__global__ void relu_kernel(float4* __restrict__ p, long n4) {
    long i = (long)blockIdx.x * blockDim.x + threadIdx.x;
    if (i < n4) {
        float4 d = p[i];
        d.x = fmaxf(d.x, 0.f);
        d.y = fmaxf(d.y, 0.f);
        d.z = fmaxf(d.z, 0.f);
        d.w = fmaxf(d.w, 0.f);
        p[i] = d;
    }
}

// ---------------------------------------------------------------------------
// Launch: zero(out) ; h1 = x@W1 ; h2 = x@W2 ; scatter1 ; scatter2 ; relu(out)
// ---------------------------------------------------------------------------
extern "C" void kernel_launch(void* const* d_in, const int* in_sizes, int n_in,
                              void* d_out, int out_size, void* d_ws, size_t ws_size,
                              hipStream_t stream) {
    const float* x    = (const float*)d_in[0];
    const float* W1   = (const float*)d_in[1];
    const float* W2   = (const float*)d_in[2];
    const float* val1 = (const float*)d_in[3];
    const float* val2 = (const float*)d_in[4];
    const int*   row1 = (const int*)d_in[5];
    const int*   col1 = (const int*)d_in[6];
    const int*   row2 = (const int*)d_in[7];
    const int*   col2 = (const int*)d_in[8];

    float* out = (float*)d_out;

    const long n_nodes = in_sizes[0] / DIM;      // 100000
    const int  n_edges = in_sizes[3];            // 3200000
    const long n_elem  = n_nodes * DIM;          // == out_size

    float* h1 = (float*)d_ws;
    float* h2 = h1 + n_elem;

    // 1) zero the accumulator (d_out)
    {
        long n4 = n_elem / 4;
        int  blk = 256;
        long grd = (n4 + blk - 1) / blk;
        zero_kernel<<<(int)grd, blk, 0, stream>>>((float4*)out, n4);
    }

    // 2) dense GEMMs: h1 = x @ W1, h2 = x @ W2   (fp32 WMMA)
    {
        dim3 grid((unsigned)(n_nodes / 32), 2);  // 2 waves/block, 2 column halves
        gemm_f32_wmma_kernel<<<grid, 64, 0, stream>>>(x, W1, h1);
        gemm_f32_wmma_kernel<<<grid, 64, 0, stream>>>(x, W2, h2);
    }

    // 3) COO scatter-accumulate into out
    {
        int waves_per_block = 8;                 // 256 threads
        int blocks = (n_edges + waves_per_block - 1) / waves_per_block;
        spmm_atomic_kernel<<<blocks, 256, 0, stream>>>(val1, row1, col1, h1, out, n_edges);
        spmm_atomic_kernel<<<blocks, 256, 0, stream>>>(val2, row2, col2, h2, out, n_edges);
    }

    // 4) ReLU in place
    {
        long n4 = n_elem / 4;
        int  blk = 256;
        long grd = (n4 + blk - 1) / blk;
        relu_kernel<<<(int)grd, blk, 0, stream>>>((float4*)out, n4);
    }
}